// EncoderStage_24515673325823
// MI455X (gfx1250) — compile-verified
//
#include <hip/hip_runtime.h>
#include <hip/hip_bf16.h>

// ---------------- problem constants (from reference setup_inputs) ------------
#define CDIM   256
#define TDIM   8
#define HDIM   32
#define WDIM   32
#define LDIM   4
#define THWN   8192          // T*H*W
#define NPTS   64            // NUM_PTS^3
#define NOFF   4
#define DSUB   64            // C / NOFF
#define MROWS  256           // B*L*N
#define KIN    768           // 3*C
#define OUT_FEATS (CDIM * THWN * LDIM)   // 8388608 floats, then 256 int32

typedef float v2f __attribute__((ext_vector_type(2)));
typedef float v8f __attribute__((ext_vector_type(8)));

// uniform_flat_indices(8,32,32,4): ti={1,3,4,6}, hi=wi={4,12,19,27}
__device__ __forceinline__ int flat_idx(int n) {
    const int ti[4] = {1, 3, 4, 6};
    const int hw[4] = {4, 12, 19, 27};
    int it = n >> 4, ih = (n >> 2) & 3, iw = n & 3;
    return (ti[it] * HDIM + hw[ih]) * WDIM + hw[iw];
}

__device__ __forceinline__ float inv_sigmoid(float x) {
    x = fminf(fmaxf(x, 0.f), 1.f);
    return logf(fmaxf(x, 1e-5f) / fmaxf(1.f - x, 1e-5f));
}
__device__ __forceinline__ float sigmoidf(float x) { return 1.f / (1.f + expf(-x)); }

// ---------------- 1. output <- beta[c] everywhere ---------------------------
__global__ void k_fill_beta(const float* __restrict__ beta, float* __restrict__ out) {
    int c = blockIdx.x;
    float v = beta[c];
    float4 v4 = make_float4(v, v, v, v);
    float4* o = reinterpret_cast<float4*>(out + (size_t)c * (THWN * LDIM));
    for (int i = threadIdx.x; i < (THWN * LDIM) / 4; i += blockDim.x) o[i] = v4;
}

// ---------------- 2. global mean over (T,H,W) per (c,l) ---------------------
__global__ void k_glob_mean(const float* __restrict__ feats, float* __restrict__ gmean) {
    int c = blockIdx.x;
    const float4* f = reinterpret_cast<const float4*>(feats + (size_t)c * (THWN * LDIM));
    float4 s = make_float4(0.f, 0.f, 0.f, 0.f);
    for (int i = threadIdx.x; i < THWN; i += blockDim.x) {
        float4 v = f[i];                       // 4 contiguous L values for one thw
        s.x += v.x; s.y += v.y; s.z += v.z; s.w += v.w;
    }
    __shared__ float4 red[256];
    red[threadIdx.x] = s;
    __syncthreads();
    for (int st = 128; st > 0; st >>= 1) {
        if (threadIdx.x < st) {
            float4 o = red[threadIdx.x + st];
            red[threadIdx.x].x += o.x; red[threadIdx.x].y += o.y;
            red[threadIdx.x].z += o.z; red[threadIdx.x].w += o.w;
        }
        __syncthreads();
    }
    if (threadIdx.x == 0) {
        const float inv = 1.f / (float)THWN;
        gmean[c * 4 + 0] = red[0].x * inv;
        gmean[c * 4 + 1] = red[0].y * inv;
        gmean[c * 4 + 2] = red[0].z * inv;
        gmean[c * 4 + 3] = red[0].w * inv;
    }
}

// ---------------- 3. build X (256 x 768): [feat_pts | pos_pts | glob] -------
__global__ void k_build_x(const float* __restrict__ feats, const float* __restrict__ pos,
                          const float* __restrict__ gmean, float* __restrict__ X) {
    int idx = blockIdx.x * blockDim.x + threadIdx.x;   // < 256*768
    int r = idx / KIN, c = idx - r * KIN;
    int l = r >> 6, n = r & 63;
    int f = flat_idx(n);
    float v;
    if (c < CDIM)            v = feats[((size_t)c * THWN + f) * LDIM + l];
    else if (c < 2 * CDIM)   v = pos[((size_t)(c - CDIM) * THWN + f) * LDIM + l];
    else                     v = gmean[(c - 2 * CDIM) * 4 + l];
    X[idx] = v;
}

// ---------------- 4. fp32 WMMA GEMM:  Out = relu(A[MxK] @ B[KxN] + bias) ----
// One wave computes one 16x16 tile using V_WMMA_F32_16X16X4_F32 over K/4 steps.
// A frag (16x4): lane<16 -> M=lane, v0:K=k0, v1:K=k0+1 ; lane>=16 -> K=k0+2/3.
// B frag (4x16): lane<16 -> N=lane, v0:K=k0, v1:K=k0+1 ; lane>=16 -> K=k0+2/3.
// C/D: vgpr j -> row j + 8*(lane/16), col lane%16.
__global__ void k_gemm_wmma(const float* __restrict__ A, const float* __restrict__ Bm,
                            const float* __restrict__ bias, float* __restrict__ Out,
                            int M, int N, int K, int do_relu) {
    int wid  = (blockIdx.x * blockDim.x + threadIdx.x) >> 5;   // uniform per wave
    int lane = threadIdx.x & 31;
    int ntn  = N >> 4;
    int mt = wid / ntn, nt = wid - mt * ntn;
    if (mt * 16 >= M) return;                                   // wave-uniform
    int half = lane >> 4, r = lane & 15;

    const float* arow = A + (size_t)(mt * 16 + r) * K;
    const float* bcol = Bm + (nt * 16 + r);
    v8f acc = {};
    for (int k0 = 0; k0 < K; k0 += 4) {
        int ka = k0 + 2 * half;
        v2f a; a.x = arow[ka];                a.y = arow[ka + 1];
        v2f b; b.x = bcol[(size_t)ka * N];    b.y = bcol[(size_t)(ka + 1) * N];
        acc = __builtin_amdgcn_wmma_f32_16x16x4_f32(
            /*neg_a=*/false, a, /*neg_b=*/false, b,
            /*c_mod=*/(short)0, acc, /*reuse_a=*/false, /*reuse_b=*/false);
    }
    int col = nt * 16 + r;
    float bv = bias[col];
    #pragma unroll
    for (int j = 0; j < 8; ++j) {
        int row = mt * 16 + j + 8 * half;
        float v = acc[j] + bv;
        if (do_relu) v = fmaxf(v, 0.f);
        Out[(size_t)row * N + col] = v;
    }
}

// ---------------- 5. heads: offset/w_o per (l,no,n), next_ind per (l,n) -----
__global__ void k_heads(const float* __restrict__ X2,
                        const float* __restrict__ Wn,  const float* __restrict__ bn,
                        const float* __restrict__ Woff, const float* __restrict__ boff,
                        const float* __restrict__ Ww,   const float* __restrict__ bw,
                        float* __restrict__ samp, float* __restrict__ w_o,
                        int* __restrict__ next_ind) {
    int tid = blockIdx.x * blockDim.x + threadIdx.x;   // < 1024 = L*No*N
    int l = tid >> 8, no = (tid >> 6) & 3, n = tid & 63;
    int r = l * 64 + n;
    const float* row = X2 + (size_t)r * CDIM;
    const float* osr = row + no * DSUB;

    float o0 = boff[0], o1 = boff[1], o2 = boff[2];
    float z0 = bw[0], z1 = bw[1];
    for (int j = 0; j < DSUB; ++j) {
        float v = osr[j];
        o0 += v * Woff[j * 3 + 0];
        o1 += v * Woff[j * 3 + 1];
        o2 += v * Woff[j * 3 + 2];
        z0 += v * Ww[j * 2 + 0];
        z1 += v * Ww[j * 2 + 1];
    }
    int f = flat_idx(n);
    float wn = (float)(f & 31) / (float)(WDIM - 1);
    float hn = (float)((f >> 5) & 31) / (float)(HDIM - 1);
    float tn = (float)(f >> 10) / (float)(TDIM - 1);
    float ist = inv_sigmoid(tn), ish = inv_sigmoid(hn), isw = inv_sigmoid(wn);

    samp[tid * 3 + 0] = sigmoidf(ist + o0) * 2.f - 1.f;
    samp[tid * 3 + 1] = sigmoidf(ish + o1) * 2.f - 1.f;
    samp[tid * 3 + 2] = sigmoidf(isw + o2) * 2.f - 1.f;
    // softmax(2)[1] == sigmoid(z1 - z0)
    w_o[tid] = sigmoidf(z1 - z0);

    if (no == 0) {
        float n0 = bn[0], n1 = bn[1], n2 = bn[2];
        for (int j = 0; j < CDIM; ++j) {
            float v = row[j];
            n0 += v * Wn[j * 3 + 0];
            n1 += v * Wn[j * 3 + 1];
            n2 += v * Wn[j * 3 + 2];
        }
        int i0 = (int)rintf(sigmoidf(ist + n0) * (float)(TDIM - 1));
        int i1 = (int)rintf(sigmoidf(ish + n1) * (float)(HDIM - 1));
        int i2 = (int)rintf(sigmoidf(isw + n2) * (float)(WDIM - 1));
        next_ind[l * 64 + n] = HDIM * WDIM * i0 + WDIM * i1 + i2;
    }
}

// ---------------- 6. trilinear sample + residual + layernorm at flat pts ----
__global__ void k_sample_ln(const float* __restrict__ feats,
                            const float* __restrict__ samp, const float* __restrict__ w_o,
                            const float* __restrict__ gamma, const float* __restrict__ beta,
                            float* __restrict__ out) {
    int blk = blockIdx.x;              // 0..255  -> (l, n)
    int l = blk >> 6, n = blk & 63;
    int c = threadIdx.x;               // channel
    int f = flat_idx(n);
    const float* fb = feats + l;       // feats[(c*THW + p)*L + l]

    float acc = fb[((size_t)c * THWN + f) * LDIM];          // residual
    #pragma unroll
    for (int no = 0; no < NOFF; ++no) {
        int p = ((l * NOFF + no) << 6) + n;
        float s0 = samp[p * 3 + 0], s1 = samp[p * 3 + 1], s2 = samp[p * 3 + 2];
        float wo = w_o[p];
        // grid channel 0 -> x (W), 1 -> y (H), 2 -> z (T)  (as in reference)
        float ix = ((s0 + 1.f) * WDIM - 1.f) * 0.5f;
        float iy = ((s1 + 1.f) * HDIM - 1.f) * 0.5f;
        float iz = ((s2 + 1.f) * TDIM - 1.f) * 0.5f;
        float x0f = floorf(ix), y0f = floorf(iy), z0f = floorf(iz);
        float fx = ix - x0f, fy = iy - y0f, fz = iz - z0f;
        int x0 = (int)x0f, y0 = (int)y0f, z0 = (int)z0f;
        float sval = 0.f;
        #pragma unroll
        for (int dz = 0; dz < 2; ++dz)
            #pragma unroll
            for (int dy = 0; dy < 2; ++dy)
                #pragma unroll
                for (int dx = 0; dx < 2; ++dx) {
                    int xi = x0 + dx, yi = y0 + dy, zi = z0 + dz;
                    bool valid = (xi >= 0) & (xi < WDIM) & (yi >= 0) & (yi < HDIM)
                               & (zi >= 0) & (zi < TDIM);
                    float w = (dx ? fx : 1.f - fx) * (dy ? fy : 1.f - fy) * (dz ? fz : 1.f - fz);
                    int cx = min(max(xi, 0), WDIM - 1);
                    int cy = min(max(yi, 0), HDIM - 1);
                    int cz = min(max(zi, 0), TDIM - 1);
                    int fc = (cz * HDIM + cy) * WDIM + cx;
                    float v = fb[((size_t)c * THWN + fc) * LDIM];
                    sval += v * (valid ? w : 0.f);
                }
        acc += wo * sval;
    }

    // layernorm over the 256 channels in this block
    __shared__ float red[256];
    red[c] = acc;
    __syncthreads();
    for (int st = 128; st > 0; st >>= 1) { if (c < st) red[c] += red[c + st]; __syncthreads(); }
    float mu = red[0] * (1.f / 256.f);
    __syncthreads();
    float dmu = acc - mu;
    red[c] = dmu * dmu;
    __syncthreads();
    for (int st = 128; st > 0; st >>= 1) { if (c < st) red[c] += red[c + st]; __syncthreads(); }
    float var = red[0] * (1.f / 256.f);
    out[((size_t)c * THWN + f) * LDIM + l] = dmu * rsqrtf(var + 1e-5f) * gamma[c] + beta[c];
}

// ---------------------------------------------------------------------------
extern "C" void kernel_launch(void* const* d_in, const int* in_sizes, int n_in,
                              void* d_out, int out_size, void* d_ws, size_t ws_size,
                              hipStream_t stream) {
    const float* feats = (const float*)d_in[0];
    const float* pos   = (const float*)d_in[1];
    const float* W1    = (const float*)d_in[2];
    const float* b1    = (const float*)d_in[3];
    const float* W2    = (const float*)d_in[4];
    const float* b2    = (const float*)d_in[5];
    const float* Wn    = (const float*)d_in[6];
    const float* bn    = (const float*)d_in[7];
    // d_in[8]=Wl, d_in[9]=bl: unused — softmax rows sum to 1, wl_sum == 1 at set points
    const float* Woff  = (const float*)d_in[10];
    const float* boff  = (const float*)d_in[11];
    const float* Ww    = (const float*)d_in[12];
    const float* bw    = (const float*)d_in[13];
    const float* gamma = (const float*)d_in[14];
    const float* beta  = (const float*)d_in[15];
    float* out = (float*)d_out;

    // workspace layout (floats): ~1.33 MB total
    float* ws    = (float*)d_ws;
    float* gmean = ws;                    // 1024
    float* X     = gmean + 1024;          // 256*768
    float* X1    = X + MROWS * KIN;       // 256*256
    float* X2    = X1 + MROWS * CDIM;     // 256*256
    float* samp  = X2 + MROWS * CDIM;     // 1024*3
    float* wo    = samp + 1024 * 3;       // 1024

    k_fill_beta<<<CDIM, 256, 0, stream>>>(beta, out);
    k_glob_mean<<<CDIM, 256, 0, stream>>>(feats, gmean);
    k_build_x<<<(MROWS * KIN) / 256, 256, 0, stream>>>(feats, pos, gmean, X);
    // 256 tiles, 8 waves/block -> 32 blocks per GEMM
    k_gemm_wmma<<<32, 256, 0, stream>>>(X,  W1, b1, X1, MROWS, CDIM, KIN,  1);
    k_gemm_wmma<<<32, 256, 0, stream>>>(X1, W2, b2, X2, MROWS, CDIM, CDIM, 1);
    k_heads<<<4, 256, 0, stream>>>(X2, Wn, bn, Woff, boff, Ww, bw, samp, wo,
                                   (int*)(out + (size_t)OUT_FEATS));
    k_sample_ln<<<MROWS, 256, 0, stream>>>(feats, samp, wo, gamma, beta, out);
}